// VanillaGNNLayer_53291954208955
// MI455X (gfx1250) — compile-verified
//
#include <hip/hip_runtime.h>
#include <hip/hip_bf16.h>

typedef __attribute__((ext_vector_type(2))) float v2f;
typedef __attribute__((ext_vector_type(8))) float v8f;

#define DIM 128  // DIM_IN == DIM_OUT == 128

// ---------------------------------------------------------------------------
// Kernel 1: zero the output accumulator (harness poisons d_out before timing)
// ---------------------------------------------------------------------------
__global__ void gnn_zero_kernel(float* __restrict__ out, int n) {
    int i = blockIdx.x * blockDim.x + threadIdx.x;
    if (i < n) out[i] = 0.0f;
}

// ---------------------------------------------------------------------------
// Kernel 2: h = x @ W^T  via V_WMMA_F32_16X16X4_F32 (full fp32 precision).
//   h[nrow, o] = sum_k x[nrow,k] * W[o,k]    (W is [dim_out, dim_in] row-major)
// Block = 256 threads = 8 waves. Wave w handles rows [blk*128 + w*16, +16)
// and the full 128-wide output (8 column tiles), reusing its A fragment
// across all 8 WMMAs per K-step.
//
// f32 WMMA VGPR layouts (ISA 7.12.2):
//   A 16x4:  lanes 0-15 -> M=lane, khalf=lane>>4 picks K-pair {2*khalf, 2*khalf+1}
//   B 4x16:  lane&15 -> N, same khalf K-pair selection (consecutive floats of W row)
//   C/D 16x16: VGPR v -> M = v + 8*khalf, N = lane&15
// ---------------------------------------------------------------------------
__global__ void __launch_bounds__(256, 2)
gnn_gemm_wmma_kernel(const float* __restrict__ x, const float* __restrict__ W,
                     float* __restrict__ h, int nNodes) {
    const int lane  = threadIdx.x & 31;
    const int wave  = threadIdx.x >> 5;        // 0..7
    const int m     = lane & 15;               // M (for A) / N (for B,C,D)
    const int khalf = lane >> 4;               // 0 or 1: selects K pair

    const int row0 = blockIdx.x * 128 + wave * 16;
    if (row0 >= nNodes) return;                // wave-uniform exit

    int rowA = row0 + m;
    if (rowA >= nNodes) rowA = nNodes - 1;     // clamp loads; stores guarded below

    const float* __restrict__ xr = x + (size_t)rowA * DIM + 2 * khalf;

    v8f acc[8] = {};                           // 8 N-tiles of 16, fp32 C/D

#pragma unroll 4
    for (int k = 0; k < DIM; k += 4) {
        v2f a = *(const v2f*)(xr + k);         // x[rowA, k+2*khalf .. +1]
#pragma unroll
        for (int t = 0; t < 8; ++t) {
            // B[kk, n] = W[n, kk]; n = t*16 + m; kk = k + 2*khalf (+1)
            const float* wp = W + (size_t)(t * 16 + m) * DIM + k + 2 * khalf;
            v2f b = *(const v2f*)wp;
            acc[t] = __builtin_amdgcn_wmma_f32_16x16x4_f32(
                /*neg_a=*/false, a, /*neg_b=*/false, b,
                /*c_mod=*/(short)0, acc[t], /*reuse_a=*/false, /*reuse_b=*/false);
        }
    }

    // Store D: VGPR v holds row (row0 + v + 8*khalf), column t*16 + m
#pragma unroll
    for (int t = 0; t < 8; ++t) {
        const int n = t * 16 + m;
#pragma unroll
        for (int v = 0; v < 8; ++v) {
            const int outM = row0 + v + 8 * khalf;
            if (outM < nNodes) h[(size_t)outM * DIM + n] = acc[t][v];
        }
    }
}

// ---------------------------------------------------------------------------
// Kernel 3: COO scatter: out[rows[e], :] += vals[e] * h[cols[e], :]
// One wave (32 lanes) per edge; each lane covers 4 consecutive floats
// (float4 L2-resident gather of h, then 4x global_atomic_add_f32).
// ---------------------------------------------------------------------------
__global__ void __launch_bounds__(256)
gnn_scatter_kernel(const float* __restrict__ h, const float* __restrict__ vals,
                   const int* __restrict__ rows, const int* __restrict__ cols,
                   float* __restrict__ out, int nEdges) {
    const int lane = threadIdx.x & 31;
    const int e = blockIdx.x * 8 + (threadIdx.x >> 5);
    if (e >= nEdges) return;

    const int   r = rows[e];
    const int   c = cols[e];
    const float v = vals[e];

    const float4 hm = ((const float4*)(h + (size_t)c * DIM))[lane];
    float* op = out + (size_t)r * DIM + lane * 4;

    unsafeAtomicAdd(op + 0, v * hm.x);   // native global_atomic_add_f32
    unsafeAtomicAdd(op + 1, v * hm.y);
    unsafeAtomicAdd(op + 2, v * hm.z);
    unsafeAtomicAdd(op + 3, v * hm.w);
}

// ---------------------------------------------------------------------------
// Kernel 4: ReLU in place
// ---------------------------------------------------------------------------
__global__ void gnn_relu_kernel(float* __restrict__ out, int n) {
    int i = blockIdx.x * blockDim.x + threadIdx.x;
    if (i < n) out[i] = fmaxf(out[i], 0.0f);
}

// ---------------------------------------------------------------------------
// Launch: inputs in setup_inputs() order: x, W, vals, rows, cols
// ---------------------------------------------------------------------------
extern "C" void kernel_launch(void* const* d_in, const int* in_sizes, int n_in,
                              void* d_out, int out_size, void* d_ws, size_t ws_size,
                              hipStream_t stream) {
    const float* x    = (const float*)d_in[0];
    const float* W    = (const float*)d_in[1];
    const float* vals = (const float*)d_in[2];
    const int*   rows = (const int*)d_in[3];
    const int*   cols = (const int*)d_in[4];

    float* out = (float*)d_out;
    float* h   = (float*)d_ws;                 // [nNodes, 128] fp32 scratch

    const int nNodes = in_sizes[0] / DIM;
    const int nEdges = in_sizes[3];
    const int nOut   = out_size;               // nNodes * DIM

    gnn_zero_kernel<<<(nOut + 255) / 256, 256, 0, stream>>>(out, nOut);

    const int gemmBlocks = (nNodes + 127) / 128;
    gnn_gemm_wmma_kernel<<<gemmBlocks, 256, 0, stream>>>(x, W, h, nNodes);

    gnn_scatter_kernel<<<(nEdges + 7) / 8, 256, 0, stream>>>(h, vals, rows, cols, out, nEdges);

    gnn_relu_kernel<<<(nOut + 255) / 256, 256, 0, stream>>>(out, nOut);
}